// GraphBranchModel_87273735454958
// MI455X (gfx1250) — compile-verified
//
#include <hip/hip_runtime.h>
#include <hip/hip_bf16.h>
#include <math.h>

// ---------------------------------------------------------------------------
// GraphBranchModel on MI455X (gfx1250, wave32).
//
// Dense projections (x@W, x@P, MLPs over N=50000 rows) use
// V_WMMA_F32_16X16X4_F32 (fp32 in / fp32 acc -> matches fp32 reference).
// 5-wave workgroups: 3125 row tiles = 625 blocks x 5 waves exactly, so no
// guards and EXEC stays all-ones around the WMMAs. The K x 16 B column slice
// is staged into LDS once per workgroup by the Tensor Data Mover
// (tensor_load_to_lds + s_wait_tensorcnt + barrier), then shared by all 5
// waves via ds_load — B global traffic /5, and the CDNA5 async-tensor path
// is exercised.
//
// Message passing (the memory-bound part, ~4.6 GB of L2-resident traffic per
// hemisphere; everything fits in the 192 MB L2) is a flat scatter: one thread
// per (edge, 4 channels), float4 gather of (x@W)[src], 4 fp32 atomics into
// agg[dst]. Self loops folded analytically with norm = dinv[i]^2.
//
// BatchNorm uses training-mode batch statistics; GCNConv bias cancels exactly
// through BN and is skipped.
// ---------------------------------------------------------------------------

#define NNODES 50000
#define NEDGES 800000
#define GEMM_WAVES 5          // 3125 row tiles = 625 blocks * 5 waves

typedef float v2f  __attribute__((ext_vector_type(2)));
typedef float v8f  __attribute__((ext_vector_type(8)));
typedef unsigned int u32x4 __attribute__((ext_vector_type(4)));
typedef int   i32x8 __attribute__((ext_vector_type(8)));
typedef int   i32x4 __attribute__((ext_vector_type(4)));

// ---------------- WMMA GEMM: C[N x M] = f(A[N x K] @ B[K x M]) --------------
// A row stride = lda, column window starts at aoff.
// val = acc (+ bias[col]); if relu: val = max(val,0); (+ add0[elem]); store.
__global__ void k_gemm_wmma(const float* __restrict__ A, int lda, int aoff,
                            const float* __restrict__ B,
                            const float* __restrict__ bias,
                            const float* __restrict__ add0,
                            float* __restrict__ C,
                            int K, int M, int relu)
{
  __shared__ float ldsB[256 * 16];       // K x 16 slice of B, up to 16 KB

  const int tid  = threadIdx.x;
  const int wave = tid >> 5;
  const int lane = tid & 31;
  const int half = lane >> 4;            // 0 / 1
  const int l16  = lane & 15;
  const int rowBase = (blockIdx.x * GEMM_WAVES + wave) * 16;
  const int colBase = blockIdx.y * 16;

  if (wave == 0) {
    // ---- TDM: DMA B[:, colBase:colBase+16] (K x 16 fp32) into LDS ----
    // D# group 0: count=1 | lds_addr | global_addr(57b) | type=2
    unsigned long long ga = (unsigned long long)(uintptr_t)(B + colBase);
    unsigned ldsOff = (unsigned)(uintptr_t)(void*)ldsB;  // low 32b of generic
                                                         // LDS ptr = byte off
    u32x4 g0;
    g0[0] = 1u;                                          // count=1
    g0[1] = ldsOff;                                      // lds_addr
    g0[2] = (unsigned)(ga & 0xffffffffull);              // global_addr[31:0]
    g0[3] = (unsigned)((ga >> 32) & 0x1ffffffull) | (2u << 30); // [56:32]|type
    // D# group 1: data_size=4B, tensor_dim0=M, tensor_dim1=K,
    //             tile_dim0=16, tile_dim1=K, tensor_dim0_stride=M
    i32x8 g1;
    g1[0] = (int)(2u << 16);                             // data_size code 2
    g1[1] = (int)(((unsigned)M & 0xffffu) << 16);        // tensor_dim0 lo16
    g1[2] = (int)(((unsigned)K & 0xffffu) << 16);        // tensor_dim1 lo16
    g1[3] = (int)(16u << 16);                            // tile_dim0 = 16
    g1[4] = K;                                           // tile_dim1 = K
    g1[5] = M;                                           // dim0_stride = M
    g1[6] = 0;
    g1[7] = 0;
    i32x4 gz4 = {0, 0, 0, 0};                            // 2-D: groups 2/3 = 0
    i32x8 gz8 = {0, 0, 0, 0, 0, 0, 0, 0};                // extra group (clang-23)
    __builtin_amdgcn_tensor_load_to_lds(g0, g1, gz4, gz4, gz8, 0);
    __builtin_amdgcn_s_wait_tensorcnt(0);
  }
  __syncthreads();                        // publish LDS tile to all 5 waves

  v8f acc = {0.f, 0.f, 0.f, 0.f, 0.f, 0.f, 0.f, 0.f};

  // A (16x4 fp32): lane l -> row (l&15), K-pair base 2*(l>>4)   [ISA 7.12.2]
  const float* Arow = A + (long)(rowBase + l16) * lda + aoff + 2 * half;
  // B (4x16 fp32) from LDS: VGPR v, lanes 0-15 -> K=v, lanes 16-31 -> K=v+2
  const float* Bl = ldsB + l16 + 2 * half * 16;

  for (int k = 0; k < K; k += 4) {
    v2f a, b;
    a.x = Arow[k + 0];
    a.y = Arow[k + 1];
    b.x = Bl[k * 16];
    b.y = Bl[k * 16 + 16];
    acc = __builtin_amdgcn_wmma_f32_16x16x4_f32(
        /*neg_a=*/false, a, /*neg_b=*/false, b,
        /*c_mod=*/(short)0, acc, /*reuse_a=*/false, /*reuse_b=*/false);
  }

  const float bv = bias ? bias[colBase + l16] : 0.f;
  // C/D (16x16 fp32): VGPR v, lanes 0-15 -> row v, lanes 16-31 -> row v+8
  #pragma unroll
  for (int v = 0; v < 8; ++v) {
    int r = rowBase + v + 8 * half;
    long idx = (long)r * M + colBase + l16;
    float val = acc[v] + bv;
    if (relu) val = fmaxf(val, 0.f);
    if (add0) val += add0[idx];
    C[idx] = val;
  }
}

// ---------------- graph prep -----------------------------------------------
__global__ void k_fill(float* __restrict__ p, float v, int n) {
  int t = blockIdx.x * blockDim.x + threadIdx.x;
  if (t < n) p[t] = v;
}

__global__ void k_degree(const int* __restrict__ dstv, float* __restrict__ deg, int e) {
  int t = blockIdx.x * blockDim.x + threadIdx.x;
  if (t < e) atomicAdd(&deg[dstv[t]], 1.0f);
}

__global__ void k_rsqrt_ip(float* __restrict__ p, int n) {
  int t = blockIdx.x * blockDim.x + threadIdx.x;
  if (t < n) p[t] = rsqrtf(p[t]);
}

__global__ void k_edge_norm(const int* __restrict__ srcv, const int* __restrict__ dstv,
                            const float* __restrict__ dinv, float* __restrict__ normv, int e) {
  int t = blockIdx.x * blockDim.x + threadIdx.x;
  if (t < e) normv[t] = dinv[srcv[t]] * dinv[dstv[t]];
}

// ---------------- aggregation ----------------------------------------------
// one thread per (edge, 4 channels); peShift = log2(M/4)
__global__ void k_edge_agg(const int* __restrict__ srcv, const int* __restrict__ dstv,
                           const float* __restrict__ normv, const float* __restrict__ xw,
                           float* __restrict__ agg, unsigned total, int M, int peShift)
{
  unsigned t = blockIdx.x * blockDim.x + threadIdx.x;
  if (t >= total) return;
  unsigned e  = t >> peShift;
  unsigned c4 = (t & ((1u << peShift) - 1u)) << 2;
  float w = normv[e];
  long sb = (long)srcv[e] * M + c4;
  long db = (long)dstv[e] * M + c4;
  const float4 v = *reinterpret_cast<const float4*>(xw + sb);
  atomicAdd(agg + db + 0, v.x * w);
  atomicAdd(agg + db + 1, v.y * w);
  atomicAdd(agg + db + 2, v.z * w);
  atomicAdd(agg + db + 3, v.w * w);
}

// self loop: agg[i,:] += xw[i,:] * dinv[i]^2   (one thread per 4 channels)
__global__ void k_self_agg(const float* __restrict__ xw, const float* __restrict__ dinv,
                           float* __restrict__ agg, int total4, int peShift)
{
  int t = blockIdx.x * blockDim.x + threadIdx.x;
  if (t >= total4) return;
  int i = t >> peShift;
  float w = dinv[i]; w = w * w;
  long b = (long)t * 4;
  float4 v = *reinterpret_cast<const float4*>(xw + b);
  float4 a = *reinterpret_cast<const float4*>(agg + b);
  a.x += v.x * w; a.y += v.y * w; a.z += v.z * w; a.w += v.w * w;
  *reinterpret_cast<float4*>(agg + b) = a;
}

// ---------------- column statistics ----------------------------------------
// blockDim.x = 64 (channels, coalesced); grid.y slabs stride over rows.
__global__ void k_colstats(const float* __restrict__ h, float* __restrict__ sum,
                           float* __restrict__ sumsq, int nrows, int M)
{
  int c = blockIdx.x * blockDim.x + threadIdx.x;
  if (c >= M) return;
  float s = 0.f, q = 0.f;
  for (int r = blockIdx.y; r < nrows; r += gridDim.y) {
    float v = h[(long)r * M + c];
    s += v; q += v * v;
  }
  atomicAdd(&sum[c], s);
  if (sumsq) atomicAdd(&sumsq[c], q);
}

__global__ void k_bn_finalize(const float* __restrict__ sum, const float* __restrict__ sumsq,
                              float* __restrict__ mu, float* __restrict__ inv, float invN)
{
  int c = threadIdx.x;
  float m = sum[c] * invN;
  float v = sumsq[c] * invN - m * m;
  mu[c]  = m;
  inv[c] = rsqrtf(v + 1e-5f);
}

// out = relu((agg-mu)*inv*gamma+beta) + xp (+ skip); mshift = log2(M/4)
__global__ void k_bn_apply(const float* __restrict__ agg, const float* __restrict__ mu,
                           const float* __restrict__ inv, const float* __restrict__ gamma,
                           const float* __restrict__ beta, const float* __restrict__ xp,
                           const float* __restrict__ skip, float* __restrict__ out,
                           int total4, int mshift)
{
  int t = blockIdx.x * blockDim.x + threadIdx.x;
  if (t >= total4) return;
  int c4 = (t & ((1 << mshift) - 1)) << 2;
  long b = (long)t * 4;
  float4 v = *reinterpret_cast<const float4*>(agg + b);
  float4 p = *reinterpret_cast<const float4*>(xp + b);
  float4 r;
  r.x = fmaxf((v.x - mu[c4 + 0]) * inv[c4 + 0] * gamma[c4 + 0] + beta[c4 + 0], 0.f) + p.x;
  r.y = fmaxf((v.y - mu[c4 + 1]) * inv[c4 + 1] * gamma[c4 + 1] + beta[c4 + 1], 0.f) + p.y;
  r.z = fmaxf((v.z - mu[c4 + 2]) * inv[c4 + 2] * gamma[c4 + 2] + beta[c4 + 2], 0.f) + p.z;
  r.w = fmaxf((v.w - mu[c4 + 3]) * inv[c4 + 3] * gamma[c4 + 3] + beta[c4 + 3], 0.f) + p.w;
  if (skip) {
    float4 s = *reinterpret_cast<const float4*>(skip + b);
    r.x += s.x; r.y += s.y; r.z += s.z; r.w += s.w;
  }
  *reinterpret_cast<float4*>(out + b) = r;
}

// ---------------- SE gating / readout / graph head --------------------------
__global__ void k_scale_cols(float* __restrict__ h, const float* __restrict__ gate,
                             int total4, int mshift)
{
  int t = blockIdx.x * blockDim.x + threadIdx.x;
  if (t >= total4) return;
  int c4 = (t & ((1 << mshift) - 1)) << 2;
  long b = (long)t * 4;
  float4 v = *reinterpret_cast<float4*>(h + b);
  v.x *= gate[c4 + 0]; v.y *= gate[c4 + 1]; v.z *= gate[c4 + 2]; v.w *= gate[c4 + 3];
  *reinterpret_cast<float4*>(h + b) = v;
}

// 64 threads, one block: gate = sigmoid(W2^T relu(W1^T mean + b1) + b2)
__global__ void k_se_gate(const float* __restrict__ colsum,
                          const float* __restrict__ w1, const float* __restrict__ b1, // 64x16
                          const float* __restrict__ w2, const float* __restrict__ b2, // 16x64
                          float* __restrict__ gate, float invN)
{
  __shared__ float mean[64];
  __shared__ float hid[16];
  int t = threadIdx.x;
  mean[t] = colsum[t] * invN;
  __syncthreads();
  if (t < 16) {
    float s = b1[t];
    for (int i = 0; i < 64; ++i) s += mean[i] * w1[i * 16 + t];
    hid[t] = fmaxf(s, 0.f);
  }
  __syncthreads();
  float s = b2[t];
  for (int j = 0; j < 16; ++j) s += hid[j] * w2[j * 64 + t];
  gate[t] = 1.f / (1.f + expf(-s));
}

__global__ void k_colmean_out(const float* __restrict__ sum, float* __restrict__ out, float invN) {
  out[threadIdx.x] = sum[threadIdx.x] * invN;
}

// 128 threads, one block: z_graph = W2^T relu(W1^T [z_l;z_r] + b1) + b2
__global__ void k_graph_mlp(const float* __restrict__ zl, const float* __restrict__ zr,
                            const float* __restrict__ w1, const float* __restrict__ b1, // 64x128
                            const float* __restrict__ w2, const float* __restrict__ b2, // 128x128
                            float* __restrict__ out)
{
  __shared__ float zin[64];
  __shared__ float hid[128];
  int t = threadIdx.x;
  if (t < 32)       zin[t] = zl[t];
  else if (t < 64)  zin[t] = zr[t - 32];
  __syncthreads();
  float s = b1[t];
  for (int i = 0; i < 64; ++i) s += zin[i] * w1[i * 128 + t];
  hid[t] = fmaxf(s, 0.f);
  __syncthreads();
  float o = b2[t];
  for (int j = 0; j < 128; ++j) o += hid[j] * w2[j * 128 + t];
  out[t] = o;
}

// ---------------------------------------------------------------------------
extern "C" void kernel_launch(void* const* d_in, const int* in_sizes, int n_in,
                              void* d_out, int out_size, void* d_ws, size_t ws_size,
                              hipStream_t stream)
{
  (void)in_sizes; (void)n_in; (void)out_size; (void)ws_size;
  const int   N    = NNODES;
  const int   E    = NEDGES;
  const float invN = 1.0f / (float)N;

  const float* x_lh = (const float*)d_in[0];
  const float* x_rh = (const float*)d_in[1];
  const int*   ei_lh = (const int*)d_in[2];
  const int*   ei_rh = (const int*)d_in[3];
  // params flattened depth-first in setup_inputs() dict insertion order:
  //  4..7   geo.l1.{w,b}, geo.l2.{w,b}
  //  8..11  morph.l1.{w,b}, morph.l2.{w,b}
  //  12+5k  blocks[k].{W,b,gamma,beta,P}   (k = 0..5)
  //  42..45 se.l1.{w,b}, se.l2.{w,b}
  //  46..49 recon.l1.{w,b}, recon.l2.{w,b}
  //  50..53 graph.l1.{w,b}, graph.l2.{w,b}
  auto P = [&](int i) { return (const float*)d_in[i]; };

  // ---- workspace layout (floats) ----
  float* ws   = (float*)d_ws;
  float* dinv = ws;  ws += N;                 // deg, then rsqrt in place
  float* nrm  = ws;  ws += E;
  float* t1   = ws;  ws += (size_t)N * 32;
  float* h0   = ws;  ws += (size_t)N * 32;
  float* h1   = ws;  ws += (size_t)N * 64;
  float* h2   = ws;  ws += (size_t)N * 128;
  float* h3   = ws;  ws += (size_t)N * 256;
  float* u2   = ws;  ws += (size_t)N * 128;
  float* u1   = ws;  ws += (size_t)N * 64;
  float* xw   = ws;  ws += (size_t)N * 256;
  float* xp   = ws;  ws += (size_t)N * 256;
  float* agg  = ws;  ws += (size_t)N * 256;
  float* ssum = ws;  ws += 256;               // ssum/ssq contiguous (one memset)
  float* ssq  = ws;  ws += 256;
  float* mu   = ws;  ws += 256;
  float* inv  = ws;  ws += 256;
  float* gate = ws;  ws += 64;

  // ---- output layout: (hv_l, rec_l, z_l, hv_r, rec_r, z_r, z_graph) ----
  float* out   = (float*)d_out;
  float* hv_l  = out;
  float* rec_l = hv_l + (size_t)N * 32;
  float* z_l   = rec_l + (size_t)N * 32;
  float* hv_r  = z_l + 32;
  float* rec_r = hv_r + (size_t)N * 32;
  float* z_r   = rec_r + (size_t)N * 32;
  float* zg    = z_r + 32;

  auto gemm = [&](const float* A, int lda, int aoff, const float* B,
                  const float* bias, const float* add0, float* C,
                  int K, int M, int relu) {
    dim3 g(N / (16 * GEMM_WAVES), M / 16), b(32 * GEMM_WAVES);
    k_gemm_wmma<<<g, b, 0, stream>>>(A, lda, aoff, B, bias, add0, C, K, M, relu);
  };

  auto run_block = [&](int blk, const float* hin, int din, int dout,
                       const float* skip, float* hout,
                       const int* srcv, const int* dstv) {
    const int base = 12 + 5 * blk;
    const float* W     = P(base + 0);
    // P(base+1) = GCN bias: cancels exactly through batch-stat BN -> skipped.
    const float* gamma = P(base + 2);
    const float* beta  = P(base + 3);
    const float* Pw    = P(base + 4);
    const int peShift  = (dout == 32) ? 3 : (dout == 64) ? 4 : (dout == 128) ? 5 : 6;
    const int total4   = N * dout / 4;

    gemm(hin, din, 0, W,  nullptr, nullptr, xw, din, dout, 0);
    gemm(hin, din, 0, Pw, nullptr, nullptr, xp, din, dout, 0);

    (void)hipMemsetAsync(agg, 0, (size_t)N * dout * sizeof(float), stream);
    {
      unsigned total = (unsigned)E << peShift;
      k_edge_agg<<<(total + 255) / 256, 256, 0, stream>>>(srcv, dstv, nrm, xw, agg,
                                                          total, dout, peShift);
    }
    k_self_agg<<<(total4 + 255) / 256, 256, 0, stream>>>(xw, dinv, agg, total4, peShift);

    (void)hipMemsetAsync(ssum, 0, 512 * sizeof(float), stream); // zero ssum+ssq
    k_colstats<<<dim3((dout + 63) / 64, 256), 64, 0, stream>>>(agg, ssum, ssq, N, dout);
    k_bn_finalize<<<1, dout, 0, stream>>>(ssum, ssq, mu, inv, invN);
    k_bn_apply<<<(total4 + 255) / 256, 256, 0, stream>>>(agg, mu, inv, gamma, beta,
                                                         xp, skip, hout, total4, peShift);
  };

  auto run_hemi = [&](const float* x, const int* ei, float* hv, float* rec, float* z) {
    const int* srcv = ei;       // edge_index[0]
    const int* dstv = ei + E;   // edge_index[1]

    // deg = 1 (self loop) + in-degree; dinv = rsqrt(deg); norm per edge
    k_fill<<<(N + 255) / 256, 256, 0, stream>>>(dinv, 1.0f, N);
    k_degree<<<(E + 255) / 256, 256, 0, stream>>>(dstv, dinv, E);
    k_rsqrt_ip<<<(N + 255) / 256, 256, 0, stream>>>(dinv, N);
    k_edge_norm<<<(E + 255) / 256, 256, 0, stream>>>(srcv, dstv, dinv, nrm, E);

    // split encoders, fuse_mode='sum'
    gemm(x,  32, 16, P(8),  P(9),  nullptr, t1, 16, 32, 1);  // morph l1 (relu)
    gemm(t1, 32, 0,  P(10), P(11), nullptr, h0, 32, 32, 0);  // morph l2 -> h0
    gemm(x,  32, 0,  P(4),  P(5),  nullptr, t1, 16, 32, 1);  // geo l1 (relu)
    gemm(t1, 32, 0,  P(6),  P(7),  h0,      h0, 32, 32, 0);  // h0 += geo l2

    run_block(0, h0, 32, 64, nullptr, h1, srcv, dstv);

    // SE channel gating on h1
    (void)hipMemsetAsync(ssum, 0, 512 * sizeof(float), stream);
    k_colstats<<<dim3(1, 256), 64, 0, stream>>>(h1, ssum, ssq, N, 64);
    k_se_gate<<<1, 64, 0, stream>>>(ssum, P(42), P(43), P(44), P(45), gate, invN);
    k_scale_cols<<<(N * 16 + 255) / 256, 256, 0, stream>>>(h1, gate, N * 16, 4);

    run_block(1, h1, 64,  128, nullptr, h2, srcv, dstv);
    run_block(2, h2, 128, 256, nullptr, h3, srcv, dstv);
    run_block(3, h3, 256, 128, h2,      u2, srcv, dstv);   // U-skip
    run_block(4, u2, 128, 64,  h1,      u1, srcv, dstv);   // U-skip (gated h1)
    run_block(5, u1, 64,  32,  h0,      hv, srcv, dstv);   // U-skip -> u0 = hv

    // recon head
    gemm(hv, 32, 0, P(46), P(47), nullptr, t1,  32, 32, 1);
    gemm(t1, 32, 0, P(48), P(49), nullptr, rec, 32, 32, 0);

    // z = u0.mean(0)
    (void)hipMemsetAsync(ssum, 0, 512 * sizeof(float), stream);
    k_colstats<<<dim3(1, 256), 64, 0, stream>>>(hv, ssum, nullptr, N, 32);
    k_colmean_out<<<1, 32, 0, stream>>>(ssum, z, invN);
  };

  run_hemi(x_lh, ei_lh, hv_l, rec_l, z_l);
  run_hemi(x_rh, ei_rh, hv_r, rec_r, z_r);
  k_graph_mlp<<<1, 128, 0, stream>>>(z_l, z_r, P(50), P(51), P(52), P(53), zg);
}